// AttentionBlock_4346506903954
// MI455X (gfx1250) — compile-verified
//
#include <hip/hip_runtime.h>

// MI455X / gfx1250, wave32. Attention collapses (single KV token => softmax==1),
// so output = x + broadcast(z[b,:]) with z from a tiny 3-GEMM chain done in
// fp32 WMMA (V_WMMA_F32_16X16X4_F32). Dominant cost is the 537MB streaming
// add (~23us at 23.3 TB/s HBM).

typedef __attribute__((ext_vector_type(2))) float v2f;
typedef __attribute__((ext_vector_type(4))) float v4f;   // native vec for NT builtins
typedef __attribute__((ext_vector_type(8))) float v8f;

#define C_DIM   256
#define B_DIM   64
#define EMB_DIM 512
#define HW_DIM  4096   // 64*64

// D(64 x 256) = A(64 x K) @ W[row_off : row_off+256, 0:K]^T + bias[bias_off:]
// One wave per 16x16 tile of D. 64 waves total = 16 blocks x 4 waves.
__global__ __launch_bounds__(128) void gemm64x256_wmma_f32(
    const float* __restrict__ A, int lda, int K,
    const float* __restrict__ W, int row_off, int ldw,
    const float* __restrict__ bias, int bias_off,
    float* __restrict__ D)
{
    const int wave = blockIdx.x * (blockDim.x >> 5) + (threadIdx.x >> 5);
    const int lane = threadIdx.x & 31;
    const int half = lane >> 4;     // 0: lanes 0-15, 1: lanes 16-31
    const int l    = lane & 15;
    const int tm   = (wave >> 4) * 16;   // 4 tiles along M (=64)
    const int tn   = (wave & 15) * 16;   // 16 tiles along N (=256)

    // A frag (16x4 f32): lane half h, VGPR j -> element A[tm+l][k0 + 2h + j]
    // B frag (4x16 f32): lane half h, VGPR j -> element W_row(tn+l)[k0 + 2h + j]
    const float* __restrict__ ap = A + (size_t)(tm + l) * lda + 2 * half;
    const float* __restrict__ bp = W + (size_t)(row_off + tn + l) * ldw + 2 * half;

    v8f acc = {};   // C/D 16x16 f32: VGPR r -> row tm + r + 8*half, col tn + l
    for (int k0 = 0; k0 < K; k0 += 4) {
        v2f a, b;
        a.x = ap[k0 + 0];
        a.y = ap[k0 + 1];
        b.x = bp[k0 + 0];
        b.y = bp[k0 + 1];
        // 8 args: (neg_a, A, neg_b, B, c_mod, C, reuse_a, reuse_b)
        acc = __builtin_amdgcn_wmma_f32_16x16x4_f32(
            false, a, false, b, (short)0, acc, false, false);
    }

    const float bv = bias[bias_off + tn + l];
#pragma unroll
    for (int r = 0; r < 8; ++r) {
        D[(size_t)(tm + r + 8 * half) * C_DIM + (tn + l)] = acc[r] + bv;
    }
}

// y[b,c,:,:] = x[b,c,:,:] + z[b,c]   (streaming, 128-bit, non-temporal)
// One block per (b,c) pair: 4096 floats = 1024 v4f; 256 threads x 4 iters.
__global__ __launch_bounds__(256) void bcast_add_f32(
    const float* __restrict__ x, const float* __restrict__ z,
    float* __restrict__ y)
{
    const int bc = blockIdx.x;             // b*C_DIM + c
    const float zv = z[bc];                // uniform per block -> scalar load
    const v4f* __restrict__ xin = (const v4f*)x + (size_t)bc * (HW_DIM / 4);
    v4f* __restrict__ yo        = (v4f*)y       + (size_t)bc * (HW_DIM / 4);
#pragma unroll
    for (int i = 0; i < 4; ++i) {
        const int idx = threadIdx.x + i * 256;
        v4f v = __builtin_nontemporal_load(&xin[idx]);
        v.x += zv; v.y += zv; v.z += zv; v.w += zv;
        __builtin_nontemporal_store(v, &yo[idx]);
    }
}

extern "C" void kernel_launch(void* const* d_in, const int* in_sizes, int n_in,
                              void* d_out, int out_size, void* d_ws, size_t ws_size,
                              hipStream_t stream) {
    // setup_inputs() order:
    const float* x         = (const float*)d_in[0];  // (64,256,64,64)
    const float* cond_emb  = (const float*)d_in[1];  // (64,512)
    // d_in[2] ln_gamma, d_in[3] ln_beta: dead code (softmax of length-1 axis)
    const float* in_proj_w = (const float*)d_in[4];  // (768,256)
    const float* in_proj_b = (const float*)d_in[5];  // (768,)
    const float* out_w     = (const float*)d_in[6];  // (256,256)
    const float* out_b     = (const float*)d_in[7];  // (256,)
    const float* kv_w      = (const float*)d_in[8];  // (512,512)
    const float* kv_b      = (const float*)d_in[9];  // (512,)
    float* y = (float*)d_out;

    float* ws   = (float*)d_ws;
    float* vin  = ws;                     // 64x256
    float* vbuf = ws + B_DIM * C_DIM;     // 64x256
    float* zbuf = ws;                     // 64x256 (vin dead after GEMM 2)

    // G1: v_in = cond_emb @ kv_w[256:512]^T + kv_b[256:]
    gemm64x256_wmma_f32<<<16, 128, 0, stream>>>(
        cond_emb, EMB_DIM, EMB_DIM, kv_w, C_DIM, EMB_DIM, kv_b, C_DIM, vin);
    // G2: v = v_in @ in_proj_w[512:768]^T + in_proj_b[512:]
    gemm64x256_wmma_f32<<<16, 128, 0, stream>>>(
        vin, C_DIM, C_DIM, in_proj_w, 2 * C_DIM, C_DIM, in_proj_b, 2 * C_DIM, vbuf);
    // G3: z = v @ out_w^T + out_b
    gemm64x256_wmma_f32<<<16, 128, 0, stream>>>(
        vbuf, C_DIM, C_DIM, out_w, 0, C_DIM, out_b, 0, zbuf);

    // y = x + broadcast(z)  — bandwidth-bound main kernel
    bcast_add_f32<<<B_DIM * C_DIM, 256, 0, stream>>>(x, zbuf, y);
}